// SimpleGDAModel_14010183320236
// MI455X (gfx1250) — compile-verified
//
#include <hip/hip_runtime.h>
#include <math.h>

// ---------------------------------------------------------------------------
// SimpleGDAModel on gfx1250 (MI455X, wave32, WMMA)
//   enc: X[32768,100] @ W1[100,256] +b relu -> H ; H @ W2[256,128] +b -> E
//   attractor: 50 gravity iters, per-token independent, register-resident
//   dec: AF @ W1[128,256] +b relu -> D2 ; D2 @ W2[256,100] +b -> recon
// GEMMs use V_WMMA_F32_16X16X4_F32 (full fp32 precision; all four GEMMs are
// bandwidth-bound at 23.3 TB/s so the f32 matrix path costs nothing and
// keeps the math bit-faithful to the fp32 reference).
// ---------------------------------------------------------------------------

typedef __attribute__((ext_vector_type(2))) float v2f;
typedef __attribute__((ext_vector_type(8))) float v8f;

#define NTOK    32768     // 4 * 8192
#define LATENT  128
#define NATT    10
#define GRAV    0.001f
#define STEP_SZ 0.01f
#define NITER   50
#define EPS_F   1e-8f

// ---------------------------------------------------------------------------
// fp32 WMMA GEMM:  C[M,N] = act(A[M,K] @ W[K,N] + bias[N])
// Compile-time K/N -> fully unrolled K loop, immediate-offset loads, no
// runtime bounds branches (N=100 uses clamp-address + cndmask select).
// Per-wave tile: 16M x 64N (4 accumulators, A frag reused 4x).
// Block: 8 waves -> 128M x 64N.
//
// Fragment layouts (ISA 7.12.2, wave32):
//  A 16x4:  lanes 0-15 hold row M=l, K={k,k+1}; lanes 16-31 hold K={k+2,k+3}
//  B 4x16:  vgpr0: lanes 0-15 row K=k  (N=l), lanes 16-31 row K=k+2
//           vgpr1: lanes 0-15 row K=k+1,      lanes 16-31 row K=k+3
//  C 16x16: vgpr i: lanes 0-15 -> M=i, lanes 16-31 -> M=8+i (N = lane&15)
// ---------------------------------------------------------------------------
template <int K, int N, bool RELU>
__global__ __launch_bounds__(256)
void gemm_wmma_f32(const float* __restrict__ A, const float* __restrict__ W,
                   const float* __restrict__ bias, float* __restrict__ C)
{
    const int lane = threadIdx.x & 31;
    const int wave = threadIdx.x >> 5;
    const int half = lane >> 4;      // 0: K pair (k,k+1)   1: K pair (k+2,k+3)
    const int l15  = lane & 15;

    const int m0 = blockIdx.x * 128 + wave * 16;
    const int n0 = blockIdx.y * 64;

    v8f c[4];
    #pragma unroll
    for (int nt = 0; nt < 4; ++nt) c[nt] = (v8f){};

    // A row base for this lane (includes the half-wave K offset)
    const float* __restrict__ Arow = A + (size_t)(m0 + l15) * K + 2 * half;

    // Per-tile W column base (includes half-wave K-row offset). For N not a
    // multiple of 64, clamp OOB columns to 0 (valid address) and zero the
    // loaded value with a select -> no exec-mask branches in the K loop.
    int  col[4];
    bool inb[4];
    const float* __restrict__ wbase[4];
    #pragma unroll
    for (int nt = 0; nt < 4; ++nt) {
        col[nt] = n0 + nt * 16 + l15;
        if constexpr (N % 64 == 0) {
            inb[nt]   = true;
            wbase[nt] = W + (size_t)(2 * half) * N + col[nt];
        } else {
            inb[nt]   = col[nt] < N;
            wbase[nt] = W + (size_t)(2 * half) * N + (inb[nt] ? col[nt] : 0);
        }
    }

    #pragma unroll
    for (int k = 0; k < K; k += 4) {
        v2f a = *(const v2f*)(Arow + k);               // global_load_b64, imm offset

        #pragma unroll
        for (int nt = 0; nt < 4; ++nt) {
            v2f b;
            b.x = wbase[nt][(size_t)k * N];            // row k   + 2*half
            b.y = wbase[nt][(size_t)(k + 1) * N];      // row k+1 + 2*half
            if constexpr (N % 64 != 0) {
                b.x = inb[nt] ? b.x : 0.0f;            // v_cndmask, branch-free
                b.y = inb[nt] ? b.y : 0.0f;
            }
            // D = A x B + C   (v_wmma_f32_16x16x4_f32)
            c[nt] = __builtin_amdgcn_wmma_f32_16x16x4_f32(
                        false, a, false, b, (short)0, c[nt], false, false);
        }
    }

    #pragma unroll
    for (int nt = 0; nt < 4; ++nt) {
        if (inb[nt]) {
            const float bv = bias[col[nt]];
            #pragma unroll
            for (int i = 0; i < 8; ++i) {
                const int row = m0 + i + 8 * half;
                float v = c[nt][i] + bv;
                if constexpr (RELU) v = v > 0.0f ? v : 0.0f;
                C[(size_t)row * N + col[nt]] = v;
            }
        }
    }
}

// ---------------------------------------------------------------------------
// Attractor core: one token per thread, 128-d position held in VGPRs for all
// 50 iterations (zero HBM traffic inside the loop). Attractors live in LDS;
// every lane reads the same address -> LDS broadcast, conflict-free b128 reads.
//   coef_a = G / (sqrt(|att_a - p|^2) + eps)^3
//   p[d]  += STEP * ( sum_a coef_a*att_a[d]  -  p[d] * sum_a coef_a )
// Last-iteration ||delta|| accumulated via block reduce + global float atomic.
// ---------------------------------------------------------------------------
__global__ __launch_bounds__(256)
void attractor_kernel(const float* __restrict__ E,
                      const float* __restrict__ att_g,
                      float* __restrict__ AF,
                      float* __restrict__ accum)
{
    __shared__ __align__(16) float att[NATT * LATENT];
    __shared__ float red[256];

    for (int i = threadIdx.x; i < NATT * LATENT; i += 256)
        att[i] = att_g[i];
    __syncthreads();

    const int t = blockIdx.x * 256 + threadIdx.x;
    const float* __restrict__ ep = E + (size_t)t * LATENT;

    float p[LATENT];
    #pragma unroll
    for (int d = 0; d < LATENT; d += 4) {
        float4 v = *(const float4*)(ep + d);
        p[d] = v.x; p[d + 1] = v.y; p[d + 2] = v.z; p[d + 3] = v.w;
    }

    float lastsq = 0.0f;

    for (int it = 0; it < NITER; ++it) {
        float coef[NATT];
        float csum = 0.0f;

        // pass 1: distances -> per-attractor force coefficients
        #pragma unroll
        for (int a = 0; a < NATT; ++a) {
            float d2 = 0.0f;
            #pragma unroll
            for (int d = 0; d < LATENT; d += 4) {
                float4 av = *(const float4*)&att[a * LATENT + d];
                float x0 = av.x - p[d];
                float x1 = av.y - p[d + 1];
                float x2 = av.z - p[d + 2];
                float x3 = av.w - p[d + 3];
                d2 += x0 * x0 + x1 * x1 + x2 * x2 + x3 * x3;
            }
            float dist = sqrtf(d2) + EPS_F;            // eps AFTER sqrt (ref)
            coef[a] = GRAV / (dist * dist * dist);
            csum += coef[a];
        }

        // pass 2: position update
        float sq = 0.0f;
        #pragma unroll
        for (int d = 0; d < LATENT; d += 4) {
            float s0 = 0.0f, s1 = 0.0f, s2 = 0.0f, s3 = 0.0f;
            #pragma unroll
            for (int a = 0; a < NATT; ++a) {
                float4 av = *(const float4*)&att[a * LATENT + d];
                s0 += coef[a] * av.x;
                s1 += coef[a] * av.y;
                s2 += coef[a] * av.z;
                s3 += coef[a] * av.w;
            }
            float dl0 = STEP_SZ * (s0 - p[d]     * csum);
            float dl1 = STEP_SZ * (s1 - p[d + 1] * csum);
            float dl2 = STEP_SZ * (s2 - p[d + 2] * csum);
            float dl3 = STEP_SZ * (s3 - p[d + 3] * csum);
            p[d]     += dl0;
            p[d + 1] += dl1;
            p[d + 2] += dl2;
            p[d + 3] += dl3;
            sq += dl0 * dl0 + dl1 * dl1 + dl2 * dl2 + dl3 * dl3;
        }
        lastsq = sq;   // value from the final iteration survives
    }

    // write attractor field
    float* __restrict__ op = AF + (size_t)t * LATENT;
    #pragma unroll
    for (int d = 0; d < LATENT; d += 4) {
        float4 v = { p[d], p[d + 1], p[d + 2], p[d + 3] };
        *(float4*)(op + d) = v;
    }

    // block reduce last-iteration ||delta||^2, atomically fold into accum
    red[threadIdx.x] = lastsq;
    __syncthreads();
    for (int s = 128; s > 0; s >>= 1) {
        if ((int)threadIdx.x < s) red[threadIdx.x] += red[threadIdx.x + s];
        __syncthreads();
    }
    if (threadIdx.x == 0) atomicAdd(accum, red[0]);
}

__global__ void zero_accum_kernel(float* a)
{
    if (threadIdx.x == 0) *a = 0.0f;
}

__global__ void finalize_kernel(const float* __restrict__ a,
                                float* __restrict__ out_change)
{
    if (threadIdx.x == 0) *out_change = sqrtf(*a);
}

// ---------------------------------------------------------------------------
extern "C" void kernel_launch(void* const* d_in, const int* in_sizes, int n_in,
                              void* d_out, int out_size, void* d_ws, size_t ws_size,
                              hipStream_t stream)
{
    const float* text       = (const float*)d_in[0];
    const float* enc_w1     = (const float*)d_in[1];
    const float* enc_b1     = (const float*)d_in[2];
    const float* enc_w2     = (const float*)d_in[3];
    const float* enc_b2     = (const float*)d_in[4];
    const float* attractors = (const float*)d_in[5];
    const float* dec_w1     = (const float*)d_in[6];
    const float* dec_b1     = (const float*)d_in[7];
    const float* dec_w2     = (const float*)d_in[8];
    const float* dec_b2     = (const float*)d_in[9];

    const int M = NTOK;   // 32768 tokens

    // output layout: [AF 32768x128][recon 32768x100][final_change]
    float* out     = (float*)d_out;
    float* AF      = out;
    float* recon   = out + (size_t)M * LATENT;
    float* change  = out + (size_t)M * LATENT + (size_t)M * 100;

    // workspace: H (reused as decoder hidden D2), E, accum
    float* H     = (float*)d_ws;                     // 32768*256
    float* E     = H + (size_t)M * 256;              // 32768*128
    float* accum = E + (size_t)M * LATENT;           // 1 float

    const dim3 blk(256);

    // encoder
    gemm_wmma_f32<100, 256, true><<<dim3(M / 128, 256 / 64), blk, 0, stream>>>(
        text, enc_w1, enc_b1, H);
    gemm_wmma_f32<256, 128, false><<<dim3(M / 128, 128 / 64), blk, 0, stream>>>(
        H, enc_w2, enc_b2, E);

    // attractor core
    zero_accum_kernel<<<1, 32, 0, stream>>>(accum);
    attractor_kernel<<<M / 256, blk, 0, stream>>>(E, attractors, AF, accum);

    // decoder
    gemm_wmma_f32<128, 256, true><<<dim3(M / 128, 256 / 64), blk, 0, stream>>>(
        AF, dec_w1, dec_b1, H);
    gemm_wmma_f32<256, 100, false><<<dim3(M / 128, (100 + 63) / 64), blk, 0, stream>>>(
        H, dec_w2, dec_b2, recon);

    finalize_kernel<<<1, 32, 0, stream>>>(accum, change);
}